// GNN_69423851372893
// MI455X (gfx1250) — compile-verified
//
#include <hip/hip_runtime.h>
#include <math.h>
#include <stdint.h>

#define N_NODES 50000
#define N_EDGES 600000
#define D_IN    256
#define D_GNN   128
#define SROW    516   // LDS row stride in floats (bank = (4*row+col)%64, conflict-free)

typedef __attribute__((ext_vector_type(2))) float v2f;
typedef __attribute__((ext_vector_type(8))) float v8f;
typedef __attribute__((ext_vector_type(4))) int   v4i;

// gfx1250 async global->LDS copy (ASYNCcnt-tracked). Prototype learned from
// the round-2 diagnostic: (global int4*, shared int4*, imm offset, imm cpol).
#if defined(__has_builtin)
#if __has_builtin(__builtin_amdgcn_global_load_async_to_lds_b128)
#define USE_ASYNC_LDS 1
#endif
#endif

typedef __attribute__((address_space(1))) v4i gv4i;   // global int4
typedef __attribute__((address_space(3))) v4i lv4i;   // LDS int4

__device__ __forceinline__ void wait_asynccnt0() {
#if defined(__has_builtin)
#if __has_builtin(__builtin_amdgcn_s_wait_asynccnt)
  __builtin_amdgcn_s_wait_asynccnt(0);
  return;
#else
  asm volatile("s_wait_asynccnt 0" ::: "memory");
  return;
#endif
#else
  asm volatile("s_wait_asynccnt 0" ::: "memory");
#endif
}

// ---------------- Kernel 1: zero aggregation workspace --------------------
__global__ void k_zero(float* __restrict__ ws, int n) {
  int i = blockIdx.x * blockDim.x + threadIdx.x;
  int stride = gridDim.x * blockDim.x;
  for (; i < n; i += stride) ws[i] = 0.0f;
}

// ---------------- Kernel 2: weight transpose Wt[n][k] = W[k][n] -----------
__global__ void k_transpose(const float* __restrict__ W, float* __restrict__ Wt) {
  int i = blockIdx.x * blockDim.x + threadIdx.x;  // over D_GNN*D_IN
  if (i >= D_GNN * D_IN) return;
  int n = i >> 8;        // Wt row   (0..127)
  int k = i & 255;       // Wt col   (0..255)
  Wt[i] = W[(size_t)k * D_GNN + n];
}

// ---------------- Kernel 3: edge scatter-add (mean numerator + degree) ----
// One wave32 per edge; lane carries 8 contiguous dims (two coalesced float4
// loads -> 1KB/wave) and 8 f32 atomics that resolve in L2 (51MB << 192MB L2).
__global__ void __launch_bounds__(256) k_scatter(
    const float* __restrict__ x,
    const int*   __restrict__ src,
    const int*   __restrict__ dst,
    float*       __restrict__ agg,
    float*       __restrict__ cnt) {
  int e = blockIdx.x * (blockDim.x >> 5) + (threadIdx.x >> 5);
  if (e >= N_EDGES) return;
  int lane = threadIdx.x & 31;
  int s = src[e];
  int d = dst[e];
  const float4* xv = (const float4*)(x + (size_t)s * D_IN);
  float4 v0 = xv[lane * 2 + 0];
  float4 v1 = xv[lane * 2 + 1];
  float* a = agg + (size_t)d * D_IN + lane * 8;
  atomicAdd(a + 0, v0.x); atomicAdd(a + 1, v0.y);
  atomicAdd(a + 2, v0.z); atomicAdd(a + 3, v0.w);
  atomicAdd(a + 4, v1.x); atomicAdd(a + 5, v1.y);
  atomicAdd(a + 6, v1.z); atomicAdd(a + 7, v1.w);
  if (lane == 0) atomicAdd(cnt + d, 1.0f);
}

// ---------------- Kernel 4: mean = agg / max(cnt, 1) (in place) -----------
__global__ void k_norm(float* __restrict__ agg, const float* __restrict__ cnt) {
  int i = blockIdx.x * blockDim.x + threadIdx.x;
  if (i >= N_NODES * D_IN) return;
  float c = cnt[i >> 8];
  agg[i] = agg[i] / fmaxf(c, 1.0f);
}

// ---------------- Kernel 5: h = mean@W_l + x@W_r + b_l via f32 WMMA -------
// Grid: 3125 node tiles (3125*16 == 50000 -> EXEC all-1s, no divergence).
// Block: 256 threads = 8 waves; wave w owns output columns [16w, 16w+16).
// A tile (16 x (256 mean | 256 x)) staged once per block into LDS via
// async global->LDS b128 copies; WMMA A fragments then come from ds_load_b64
// and B fragments from the pre-transposed weights as single b64 loads.
__global__ void __launch_bounds__(256) k_sage_wmma(
    const float* __restrict__ mean,
    const float* __restrict__ x,
    const float* __restrict__ Wtl,   // [128][256] transposed
    const float* __restrict__ Wtr,   // [128][256] transposed
    const float* __restrict__ bl,
    float*       __restrict__ h) {
  __shared__ float smem[16 * SROW];   // row r: [0,256)=mean, [256,512)=x, pad 4

  const int tile  = blockIdx.x;
  const int row0  = tile * 16;
  const int tid   = threadIdx.x;
  const int lane  = tid & 31;
  const int wave  = tid >> 5;
  const int n0    = wave * 16;
  const int mcol  = lane & 15;        // A row / B,C column within tile
  const int khalf = (lane >> 4) * 2;  // 0 or 2

  // ---- Stage A tile: 2 x 16KB via b128 async copies across 256 lanes
#pragma unroll
  for (int j = 0; j < 4; ++j) {
    int f4 = j * 256 + tid;           // float4 index 0..1023 per matrix
    int r  = f4 >> 6;                 // row   0..15
    int c4 = f4 & 63;                 // col/4 0..63
    const float* gm = mean + (size_t)(row0 + r) * D_IN + c4 * 4;
    const float* gx = x    + (size_t)(row0 + r) * D_IN + c4 * 4;
    float* lm = smem + r * SROW + c4 * 4;
    float* lx = smem + r * SROW + 256 + c4 * 4;
#ifdef USE_ASYNC_LDS
    __builtin_amdgcn_global_load_async_to_lds_b128(
        (gv4i*)(uintptr_t)gm, (lv4i*)(uintptr_t)lm, 0, 0);
    __builtin_amdgcn_global_load_async_to_lds_b128(
        (gv4i*)(uintptr_t)gx, (lv4i*)(uintptr_t)lx, 0, 0);
#else
    *(float4*)lm = *(const float4*)gm;
    *(float4*)lx = *(const float4*)gx;
#endif
  }
#ifdef USE_ASYNC_LDS
  wait_asynccnt0();
#endif
  __syncthreads();

  const float* Arow  = smem + mcol * SROW;                   // mean half
  const float* BlRow = Wtl + (size_t)(n0 + mcol) * D_IN;     // K-contiguous
  const float* BrRow = Wtr + (size_t)(n0 + mcol) * D_IN;

  v8f acc = {};

  // mean @ W_l
#pragma unroll 8
  for (int k0 = 0; k0 < D_IN; k0 += 4) {
    const int k = k0 + khalf;
    float2 av = *(const float2*)(Arow + k);        // ds_load_b64
    float2 bv = *(const float2*)(BlRow + k);       // global_load_b64
    v2f a; a.x = av.x; a.y = av.y;
    v2f b; b.x = bv.x; b.y = bv.y;
    acc = __builtin_amdgcn_wmma_f32_16x16x4_f32(
        false, a, false, b, (short)0, acc, false, false);
  }
  // x @ W_r (accumulate into same fragment)
#pragma unroll 8
  for (int k0 = 0; k0 < D_IN; k0 += 4) {
    const int k = k0 + khalf;
    float2 av = *(const float2*)(Arow + 256 + k);
    float2 bv = *(const float2*)(BrRow + k);
    v2f a; a.x = av.x; a.y = av.y;
    v2f b; b.x = bv.x; b.y = bv.y;
    acc = __builtin_amdgcn_wmma_f32_16x16x4_f32(
        false, a, false, b, (short)0, acc, false, false);
  }

  const float bias  = bl[n0 + mcol];
  const int   mbase = (lane >> 4) * 8;   // C: VGPR i -> M = mbase+i, N = mcol
#pragma unroll
  for (int i = 0; i < 8; i++) {
    h[(size_t)(row0 + mbase + i) * D_GNN + n0 + mcol] = acc[i] + bias;
  }
}

// ---------------- Kernel 6: pred = sigmoid(h @ W_fc + b_fc) ---------------
__global__ void k_head(const float* __restrict__ h,
                       const float* __restrict__ Wfc,
                       const float* __restrict__ bfc,
                       float*       __restrict__ pred) {
  int n = blockIdx.x * blockDim.x + threadIdx.x;
  if (n >= N_NODES) return;
  const float4* hv = (const float4*)(h + (size_t)n * D_GNN);
  const float4* wv = (const float4*)Wfc;
  float acc = 0.0f;
#pragma unroll
  for (int i = 0; i < D_GNN / 4; i++) {
    float4 hh = hv[i];
    float4 ww = wv[i];
    acc += hh.x * ww.x + hh.y * ww.y + hh.z * ww.z + hh.w * ww.w;
  }
  float z = acc + bfc[0];
  pred[n] = 1.0f / (1.0f + __expf(-z));
}

// ---------------- Host launcher -------------------------------------------
extern "C" void kernel_launch(void* const* d_in, const int* in_sizes, int n_in,
                              void* d_out, int out_size, void* d_ws, size_t ws_size,
                              hipStream_t stream) {
  const float* x   = (const float*)d_in[0];               // [50000,256]
  const int*   ei  = (const int*)  d_in[1];               // [2,600000]
  const float* Wl  = (const float*)d_in[2];               // [256,128]
  const float* bl  = (const float*)d_in[3];               // [128]
  const float* Wr  = (const float*)d_in[4];               // [256,128]
  const float* Wfc = (const float*)d_in[5];               // [128,1]
  const float* bfc = (const float*)d_in[6];               // [1]

  const int* src = ei;                                    // row 0
  const int* dst = ei + N_EDGES;                          // row 1

  float* agg = (float*)d_ws;                              // [50000,256]
  float* cnt = agg + (size_t)N_NODES * D_IN;              // [50000]
  float* Wtl = cnt + N_NODES;                             // [128,256]
  float* Wtr = Wtl + (size_t)D_GNN * D_IN;                // [128,256]

  float* h    = (float*)d_out;                            // [50000,128]
  float* pred = h + (size_t)N_NODES * D_GNN;              // [50000]

  const int zero_n = N_NODES * D_IN + N_NODES;            // agg + cnt contiguous
  k_zero<<<2048, 256, 0, stream>>>(agg, zero_n);

  const int wblk = (D_GNN * D_IN + 255) / 256;
  k_transpose<<<wblk, 256, 0, stream>>>(Wl, Wtl);
  k_transpose<<<wblk, 256, 0, stream>>>(Wr, Wtr);

  k_scatter<<<(N_EDGES + 7) / 8, 256, 0, stream>>>(x, src, dst, agg, cnt);

  k_norm<<<(N_NODES * D_IN + 255) / 256, 256, 0, stream>>>(agg, cnt);

  k_sage_wmma<<<N_NODES / 16, 256, 0, stream>>>(agg, x, Wtl, Wtr, bl, h);

  k_head<<<(N_NODES + 255) / 256, 256, 0, stream>>>(h, Wfc, bfc, pred);
}